// MHA_79791902425555
// MI455X (gfx1250) — compile-verified
//
#include <hip/hip_runtime.h>
#include <hip/hip_bf16.h>

// ---------------------------------------------------------------------------
// Decode-step MHA, fp32, bandwidth-bound (~1.27 GB traffic -> ~55us @23.3TB/s)
//   N_HEAD=32 HEAD_DIM=128 D=4096 B=8 NUM_PAGES=32 PAGE=128
// ---------------------------------------------------------------------------

#define NHEAD   32
#define HDIM    128
#define DMODEL  4096
#define BATCH   8
#define NPAGES  32
#define PAGESZ  128
#define D3      (3 * DMODEL)          // 12288

typedef __attribute__((ext_vector_type(2))) float v2f;
typedef __attribute__((ext_vector_type(4))) float f4;
typedef __attribute__((ext_vector_type(8))) float v8f;

#define ATTN_SCALE 0.08838834764831845f   // 1/sqrt(128)

__device__ __forceinline__ float bcast_lane(float v, int src) {
  return __uint_as_float(__builtin_amdgcn_readlane(__float_as_uint(v), src));
}

// ---------------------------------------------------------------------------
// Kernel 0: initialize accumulators & padded operands.
//   qkv <- b_attn broadcast, out <- b_proj broadcast, num/den <- 0,
//   xpad[16,4096] <- {x rows 0..7, zeros rows 8..15}, ypad rows 8..15 <- 0.
// Re-run every call (harness poisons buffers once, atomics add on top).
// ---------------------------------------------------------------------------
__global__ __launch_bounds__(256) void init_kernel(
    const float* __restrict__ x, const float* __restrict__ b_attn,
    const float* __restrict__ b_proj, float* __restrict__ qkv,
    float* __restrict__ num, float* __restrict__ den,
    float* __restrict__ xpad, float* __restrict__ ypad,
    float* __restrict__ out) {
  int i = blockIdx.x * blockDim.x + threadIdx.x;   // up to 98304
  if (i < BATCH * D3)      qkv[i] = b_attn[i % D3];
  if (i < BATCH * DMODEL)  { num[i] = 0.0f; out[i] = b_proj[i % DMODEL]; }
  if (i < BATCH * NHEAD)   den[i] = 0.0f;
  if (i < 16 * DMODEL) {
    int row = i >> 12;                             // /4096
    xpad[i] = (row < BATCH) ? x[i] : 0.0f;         // rows 0..7 contiguous copy
  }
  if (i < BATCH * DMODEL)  ypad[BATCH * DMODEL + i] = 0.0f;  // ypad rows 8..15
}

// ---------------------------------------------------------------------------
// WMMA f32 16x16x4 GEMM:  C[8,N] += A[16,K] * W[K,N]   (A rows 8..15 are zero)
// One wave = one 16-column tile x one K-chunk (K/KSPLIT). Both W streams are
// indexed by independent 32-bit element offsets (u32 fits: W <= 201MB) so the
// backend keeps SADDR-base + voffset (scale_offset) addressing; A uses one
// pointer with immediate offsets. 1 NT b32-pair of W + 1 WMMA per K=4 step.
// A-fragment layout per ISA 7.12.2 (16x4 f32): lanes0-15 K={0,1}, lanes16-31
// K={2,3}. B assumed analogous (VGPR0={K0,K1}, VGPR1={K2,K3} per half-wave).
// ---------------------------------------------------------------------------
#define KSPLIT 4

__global__ __launch_bounds__(256) void gemm8_wmma(
    const float* __restrict__ A,   // [16, K], rows 8..15 zero
    const float* __restrict__ W,   // [K, N] row-major
    float* __restrict__ C,         // [8, N] pre-initialized with bias
    int K, int N) {
  const int lane  = threadIdx.x & 31;
  const int gwave = blockIdx.x * (blockDim.x >> 5) + (threadIdx.x >> 5);
  const int ntiles = N >> 4;
  const int tile   = gwave % ntiles;
  const int kchunk = gwave / ntiles;
  const int klen   = K / KSPLIT;
  const int k0     = kchunk * klen;
  const int col0   = tile << 4;

  const int mrow = lane & 15;            // A row / B column this lane feeds
  const int koff = (lane >> 4) << 1;     // 0 (lanes0-15) or 2 (lanes16-31)
  const int r01  = (lane >> 4);          // B: row parity per half-wave

  const float* ap = A + (size_t)mrow * K + k0 + koff;
  // two independent u32 element-offset induction vars -> both W loads stay in
  // scalar-base + 32-bit voffset form (no 64-bit address math in the loop)
  unsigned woff  = (unsigned)(k0 + r01) * (unsigned)N + (unsigned)(col0 + mrow);
  unsigned woff2 = woff + 2u * (unsigned)N;   // rows k+2 / k+3
  const unsigned wstep = 4u * (unsigned)N;    // +4 K-rows per WMMA

  v8f acc = {};
#pragma unroll 8
  for (int it = 0; it < klen / 4; ++it) {     // one WMMA (K=4) per trip
    v2f a; a.x = ap[0]; a.y = ap[1];          // merges to global_load_b64
    v2f b; b.x = __builtin_nontemporal_load(W + woff);
           b.y = __builtin_nontemporal_load(W + woff2);
    acc = __builtin_amdgcn_wmma_f32_16x16x4_f32(false, a, false, b,
                                                (short)0, acc, false, false);
    ap    += 4;
    woff  += wstep;
    woff2 += wstep;
  }
  // C layout: VGPR v, lanes0-15 -> row v (rows 0..7 = valid batch rows)
  if (lane < 16) {
#pragma unroll
    for (int v = 0; v < 8; ++v)
      atomicAdd(&C[(size_t)v * N + col0 + lane], acc[v]);
  }
}

// ---------------------------------------------------------------------------
// Attention partials: one block (128 thr = 4 waves) per (b, h, page-group);
// one wave per page. 32 keys per wave per step, two phases:
//   phase 1: lane l owns key l -- full 128-dim dot vs q staged in LDS
//            (uniform-address ds_load_b128 broadcast); one exp per lane;
//            denominator accumulated here (1 add per 32 keys per lane).
//   phase 2: v_readlane broadcasts score j; FMA coalesced 512B V row j.
// num = sum exp(s)*v, den = sum exp(s)  (telescoped reference update).
// ---------------------------------------------------------------------------
__global__ __launch_bounds__(128) void attn_partial(
    const float* __restrict__ qkv,                  // [8, 12288]
    const float* __restrict__ kc,                   // [32,8,32,128,128]
    const float* __restrict__ vc,
    float* __restrict__ num,                        // [8,32,128]
    float* __restrict__ den) {                      // [8,32]
  __shared__ float sq[HDIM];
  const int lane = threadIdx.x & 31;
  const int wv   = threadIdx.x >> 5;                // 0..3
  int idx = blockIdx.x;                             // 0..2047
  const int grp = idx & 7;  idx >>= 3;
  const int h   = idx & 31; idx >>= 5;
  const int b   = idx;                              // 0..7
  const int page = grp * 4 + wv;

  // stage q[b,h,:] once per block (all 4 waves share (b,h))
  sq[threadIdx.x] = qkv[(size_t)b * D3 + h * HDIM + threadIdx.x];
  __syncthreads();

  const size_t base = ((((size_t)page * BATCH + b) * NHEAD + h) * PAGESZ) * HDIM;
  const float* kp = kc + base;
  const float* vp = vc + base;

  f4 acc = {};
  float dacc = 0.0f;
  for (int kb = 0; kb < PAGESZ; kb += 32) {
    // speculative lookahead into the next 32-key block
    __builtin_prefetch(kp + (size_t)(kb + 32 + lane) * HDIM, 0, 0);
    __builtin_prefetch(vp + (size_t)(kb + 32 + lane) * HDIM, 0, 0);

    // ---- phase 1: score for key (kb+lane) ----
    const float* krow = kp + (size_t)(kb + lane) * HDIM;
    float s = 0.0f;
#pragma unroll
    for (int i = 0; i < 32; ++i) {
      f4 kr = __builtin_nontemporal_load((const f4*)(krow + 4 * i));
      f4 qi = *(const f4*)(sq + 4 * i);   // LDS broadcast (uniform address)
      s += qi.x * kr.x + qi.y * kr.y + qi.z * kr.z + qi.w * kr.w;
    }
    const float p = __expf(s * ATTN_SCALE);
    dacc += p;                            // den partial, own key only

    // ---- phase 2: accumulate p_j * V[j, 4l..4l+3] ----
    const float* vrow = vp + (size_t)kb * HDIM + lane * 4;
#pragma unroll
    for (int j = 0; j < 32; ++j) {
      const float pj = bcast_lane(p, j);
      f4 vr = __builtin_nontemporal_load((const f4*)(vrow + (size_t)j * HDIM));
      acc.x += pj * vr.x; acc.y += pj * vr.y;
      acc.z += pj * vr.z; acc.w += pj * vr.w;
    }
  }
  float* np = num + ((size_t)b * NHEAD + h) * HDIM + lane * 4;
  atomicAdd(np + 0, acc.x); atomicAdd(np + 1, acc.y);
  atomicAdd(np + 2, acc.z); atomicAdd(np + 3, acc.w);
  // wave-reduce the per-lane denominator partials once per page
#pragma unroll
  for (int off = 16; off > 0; off >>= 1)
    dacc += __shfl_xor(dacc, off, 32);
  if (lane == 0) atomicAdd(den + b * NHEAD + h, dacc);
}

// ---------------------------------------------------------------------------
// Finalize y: add current-token (k_cur, v_cur) contribution, divide by denom.
// One wave per (b, h). Writes rows 0..7 of ypad (rows 8..15 zeroed by init).
// ---------------------------------------------------------------------------
__global__ __launch_bounds__(32) void finalize_y(
    const float* __restrict__ qkv, const float* __restrict__ num,
    const float* __restrict__ den, float* __restrict__ ypad) {
  const int lane = threadIdx.x;
  const int h = blockIdx.x & 31;
  const int b = blockIdx.x >> 5;
  const float* qp = qkv + (size_t)b * D3 + h * HDIM;
  const f4 q4 = *(const f4*)(qp + lane * 4);
  const f4 k4 = *(const f4*)(qp + DMODEL + lane * 4);       // k_cur
  const f4 v4 = *(const f4*)(qp + 2 * DMODEL + lane * 4);   // v_cur
  float part = q4.x * k4.x + q4.y * k4.y + q4.z * k4.z + q4.w * k4.w;
#pragma unroll
  for (int off = 16; off > 0; off >>= 1)
    part += __shfl_xor(part, off, 32);
  const float p = __expf(part * ATTN_SCALE);
  const f4 n4 = *(const f4*)(num + ((size_t)b * NHEAD + h) * HDIM + lane * 4);
  const float dd = den[b * NHEAD + h] + p;
  f4 o;
  o.x = (n4.x + p * v4.x) / dd;
  o.y = (n4.y + p * v4.y) / dd;
  o.z = (n4.z + p * v4.z) / dd;
  o.w = (n4.w + p * v4.w) / dd;
  *(f4*)(ypad + (size_t)b * DMODEL + h * HDIM + lane * 4) = o;
}

// ---------------------------------------------------------------------------
extern "C" void kernel_launch(void* const* d_in, const int* in_sizes, int n_in,
                              void* d_out, int out_size, void* d_ws, size_t ws_size,
                              hipStream_t stream) {
  (void)in_sizes; (void)n_in; (void)out_size; (void)ws_size;
  const float* x      = (const float*)d_in[0];   // [8,1,4096]
  const float* kcache = (const float*)d_in[1];   // [32,8,32,128,128]
  const float* vcache = (const float*)d_in[2];
  const float* w_attn = (const float*)d_in[3];   // [4096,12288]
  const float* b_attn = (const float*)d_in[4];   // [12288]
  const float* w_proj = (const float*)d_in[5];   // [4096,4096]
  const float* b_proj = (const float*)d_in[6];   // [4096]
  float* out = (float*)d_out;                    // [8,1,4096]

  // workspace layout (floats)
  float* ws   = (float*)d_ws;
  float* qkv  = ws;                              // 8*12288  = 98304
  float* num  = qkv + BATCH * D3;                // 8*32*128 = 32768
  float* den  = num + BATCH * NHEAD * HDIM;      // 256
  float* xpad = den + BATCH * NHEAD;             // 16*4096  = 65536
  float* ypad = xpad + 16 * DMODEL;              // 16*4096  = 65536

  // 1) init accumulators / biases / zero-padded operands
  init_kernel<<<(BATCH * D3 + 255) / 256, 256, 0, stream>>>(
      x, b_attn, b_proj, qkv, num, den, xpad, ypad, out);

  // 2) qkv projection: N=12288 -> 768 tiles * KSPLIT=4 -> 3072 waves / 384 blocks
  gemm8_wmma<<<(D3 / 16) * KSPLIT / 8, 256, 0, stream>>>(
      xpad, w_attn, qkv, DMODEL, D3);

  // 3) attention partials over the 1 GB KV stream: 8*32*8 = 2048 blocks
  attn_partial<<<BATCH * NHEAD * 8, 128, 0, stream>>>(qkv, kcache, vcache, num, den);

  // 4) finalize y with current-token contribution
  finalize_y<<<BATCH * NHEAD, 32, 0, stream>>>(qkv, num, den, ypad);

  // 5) output projection: N=4096 -> 256 tiles * 4 -> 1024 waves / 128 blocks
  gemm8_wmma<<<(DMODEL / 16) * KSPLIT / 8, 256, 0, stream>>>(
      ypad, w_proj, out, DMODEL, DMODEL);
}